// Gemma4VisionPooler_2035814498747
// MI455X (gfx1250) — compile-verified
//
#include <hip/hip_runtime.h>
#include <hip/hip_bf16.h>
#include <stdint.h>

typedef __attribute__((ext_vector_type(2))) float v2f;
typedef __attribute__((ext_vector_type(8))) float v8f;

// Problem constants (from the reference)
constexpr int kB     = 16;    // batch
constexpr int kGrid  = 48;    // 48x48 patch grid
constexpr int kNP    = kGrid * kGrid;   // 2304 patches
constexpr int kH     = 1152;  // hidden size
constexpr int kPool  = 3;     // 3x3 pooling
constexpr int kBinsX = kGrid / kPool;   // 16 bins per row
constexpr int kBins  = kBinsX * kBinsX; // 256 bins
constexpr int kSlab  = kPool * kGrid;   // 144 patches feeding one bin-row
constexpr int kKSteps = kSlab / 4;      // 36 WMMA K-steps (K=4 each)
constexpr int kNTiles = kH / 16;        // 72 channel tiles of 16
constexpr int kWaves  = 8;              // 256-thread block = 8 wave32

__global__ __launch_bounds__(256, 2)
void gemma4_pooler_wmma(const float* __restrict__ hs,
                        const int* __restrict__ pos,
                        const uint8_t* __restrict__ pad,
                        float* __restrict__ outPooled,
                        float* __restrict__ outMask)
{
    const int b    = (int)blockIdx.x >> 4;   // batch
    const int by   = (int)blockIdx.x & 15;   // bin row
    const int tid  = (int)threadIdx.x;
    const int lane = tid & 31;
    const int wave = tid >> 5;

    __shared__ float s_w[kSlab];   // per-slab-patch weight: valid * sqrt(H)/count
    __shared__ int   s_bx[kSlab];  // per-slab-patch bin-x (0..15)
    __shared__ int   s_cnt[kBinsX];

    const size_t slabBase = (size_t)b * kNP + (size_t)by * kSlab;

    // --- per-bin valid counts + mask output (threads 0..15) ---
    if (tid < kBinsX) {
        int cnt = 0;
        #pragma unroll
        for (int yl = 0; yl < kPool; ++yl)
            #pragma unroll
            for (int dx = 0; dx < kPool; ++dx)
                cnt += (pad[slabBase + (size_t)yl * kGrid + tid * kPool + dx] == 0) ? 1 : 0;
        s_cnt[tid] = cnt;
        outMask[(size_t)b * kBins + by * kBinsX + tid] = (cnt > 0) ? 1.0f : 0.0f;
    }
    __syncthreads();

    // --- per-patch weights from position ids + padding (threads 0..143) ---
    if (tid < kSlab) {
        const size_t gp = slabBase + tid;
        int px = pos[gp * 2 + 0];
        px = px < 0 ? 0 : px;
        int bx = px / kPool;
        bx = bx > (kBinsX - 1) ? (kBinsX - 1) : bx;
        const bool valid = (pad[gp] == 0);
        const int  cnt   = s_cnt[bx];
        const float rootH = 33.94112549695428f;  // sqrt(1152)
        s_bx[tid] = bx;
        s_w[tid]  = (valid && cnt > 0) ? (rootH / (float)cnt) : 0.0f;
    }
    __syncthreads();

    // A-fragment per ISA layout: lane L<16 holds M=L, K={4t,4t+1};
    // lanes 16..31 hold the same M=L-16 with K={4t+2,4t+3}.
    const int m   = lane & 15;         // A: bin index; B/D: column (channel)
    const int khi = (lane >> 4) << 1;  // 0 for lanes 0-15, 2 for lanes 16-31

    // --- Precompute ALL A fragments once per wave (independent of channel tile).
    //     Loads are unconditional into locals, select is a v_cndmask, result
    //     lives in registers and is reused across the 9 channel tiles.
    v2f aFrag[kKSteps];
    #pragma unroll
    for (int t = 0; t < kKSteps; ++t) {
        const int k0  = (t << 2) + khi;
        const float w0 = s_w[k0];
        const float w1 = s_w[k0 + 1];
        const int  bx0 = s_bx[k0];
        const int  bx1 = s_bx[k0 + 1];
        v2f a;
        a.x = (bx0 == m) ? w0 : 0.0f;
        a.y = (bx1 == m) ? w1 : 0.0f;
        aFrag[t] = a;
    }

    // --- WMMA: pooled(16 bins x 16 chans) += A(16x4) * B(4 patches x 16 chans),
    //     K-reduction over the 144-patch slab in 36 steps. B loads are 64B-coalesced
    //     per half-wave, every hs element is touched exactly once -> non-temporal.
    for (int nt = wave; nt < kNTiles; nt += kWaves) {
        const int chan = nt << 4;
        const float* __restrict__ hsb = hs + slabBase * kH + chan + m;

        // pull the next tile this wave will touch toward the caches
        __builtin_prefetch(hsb + (size_t)kWaves * 16, 0, 1);

        v8f acc = {0.f, 0.f, 0.f, 0.f, 0.f, 0.f, 0.f, 0.f};

        #pragma unroll
        for (int t = 0; t < kKSteps; ++t) {
            const int k0 = (t << 2) + khi;
            v2f bb;
            bb.x = __builtin_nontemporal_load(&hsb[(size_t)k0 * kH]);
            bb.y = __builtin_nontemporal_load(&hsb[(size_t)(k0 + 1) * kH]);
            acc = __builtin_amdgcn_wmma_f32_16x16x4_f32(
                false, aFrag[t], false, bb, (short)0, acc, false, false);
        }

        // D layout: VGPR v -> row v + 8*(lane>=16), col = lane&15
        float* __restrict__ op =
            outPooled + ((size_t)b * kBins + (size_t)by * kBinsX) * kH + chan + m;
        const int mrow = (lane >> 4) << 3;
        #pragma unroll
        for (int v = 0; v < 8; ++v)
            __builtin_nontemporal_store(acc[v], &op[(size_t)(mrow + v) * kH]);
    }
}

extern "C" void kernel_launch(void* const* d_in, const int* in_sizes, int n_in,
                              void* d_out, int out_size, void* d_ws, size_t ws_size,
                              hipStream_t stream) {
    const float*   hs  = (const float*)d_in[0];
    const int*     pos = (const int*)d_in[1];
    const uint8_t* pad = (const uint8_t*)d_in[2];
    // d_in[3] = output_length scalar (256), known statically here.

    float* outPooled = (float*)d_out;
    float* outMask   = outPooled + (size_t)kB * kBins * kH;  // mask appended flat

    dim3 grid(kB * kBinsX);  // one block per (batch, bin-row) = 256 blocks
    dim3 block(256);         // 8 wave32
    hipLaunchKernelGGL(gemma4_pooler_wmma, grid, block, 0, stream,
                       hs, pos, pad, outPooled, outMask);
}